// LSTMNetwork_78494822301814
// MI455X (gfx1250) — compile-verified
//
#include <hip/hip_runtime.h>
#include <math.h>

#define BATCH 64
#define FREQ 128
#define TSEQ 256
#define HID 1024
#define OUTD 8
#define ZDIM (FREQ + HID)   // 1152
#define G4 (4 * HID)        // 4096
#define LN_EPS 1e-5f

typedef float v2f __attribute__((ext_vector_type(2)));
typedef float v8f __attribute__((ext_vector_type(8)));

__device__ __forceinline__ float sigm(float x) { return 1.0f / (1.0f + expf(-x)); }

// ---------------------------------------------------------------------------
// x [B, FREQ, T] -> xT [T, B, FREQ]  (unit-stride K loads for the A fragment)
// ---------------------------------------------------------------------------
__global__ void transpose_x_kernel(const float* __restrict__ x,
                                   float* __restrict__ xT) {
  int o = blockIdx.x * 256 + threadIdx.x;       // exact multiple, no guard
  int f = o & (FREQ - 1);
  int b = (o >> 7) & (BATCH - 1);
  int t = o >> 13;
  xT[o] = x[(b * FREQ + f) * TSEQ + t];
}

// ---------------------------------------------------------------------------
// Fuse + transpose the 4 gate weight matrices once:
//   Wt[n][k] = Wg[k][n%1024],  n = 0..4095 over [Wf|Wi|Wc|Wo], k = 0..1151
// so a lane's WMMA B-fragment (two consecutive K) is one aligned b64 load.
// LDS-tiled 32x32 transpose, both global accesses fully coalesced.
// ---------------------------------------------------------------------------
__global__ __launch_bounds__(256) void transpose_w_kernel(
    const float* __restrict__ Wf, const float* __restrict__ Wi,
    const float* __restrict__ Wc, const float* __restrict__ Wo,
    float* __restrict__ Wt) {
  __shared__ float tile[32][33];
  const int g = blockIdx.z;
  const float* __restrict__ W = (g == 0) ? Wf : (g == 1) ? Wi : (g == 2) ? Wc : Wo;
  const int c0 = blockIdx.x * 32;               // column tile in [0,1024)
  const int k0 = blockIdx.y * 32;               // row tile in [0,1152)
  const int tx = threadIdx.x & 31;
  const int ty = threadIdx.x >> 5;              // 0..7
  #pragma unroll
  for (int i = 0; i < 4; i++)
    tile[ty + 8 * i][tx] = W[(size_t)(k0 + ty + 8 * i) * HID + (c0 + tx)];
  __syncthreads();
  #pragma unroll
  for (int i = 0; i < 4; i++)
    Wt[(size_t)(g * HID + c0 + ty + 8 * i) * ZDIM + (k0 + tx)] = tile[tx][ty + 8 * i];
}

__global__ void zero_kernel(float* __restrict__ p, int n) {
  int i = blockIdx.x * 256 + threadIdx.x;
  if (i < n) p[i] = 0.0f;
}

// ---------------------------------------------------------------------------
// gates[64,4096] = concat(x_t[64,128], h[64,1024]) @ W_gates(1152,4096)
// fp32 WMMA (V_WMMA_F32_16X16X4_F32), K=1152. Each wave computes two 16x16
// N tiles sharing one A fragment: 3 b64 loads per 2 WMMAs.
// grid (4, 16), block 256 (8 waves) -> 512 waves.
// ---------------------------------------------------------------------------
__global__ __launch_bounds__(256) void gate_gemm_kernel(
    const float* __restrict__ xT, const float* __restrict__ h,
    const float* __restrict__ Wt, float* __restrict__ gates, int t) {
  const int lane  = threadIdx.x & 31;
  const int wave  = threadIdx.x >> 5;
  const int m     = blockIdx.x;                 // 0..3  (M tiles of 16)
  const int strip = blockIdx.y * 8 + wave;      // 0..127 (strips of 32 cols)
  const int n0    = strip * 32;

  const int row = m * 16 + (lane & 15);
  const int khi = (lane >> 4) << 1;             // lanes 0-15: K+{0,1}; 16-31: K+{2,3}

  const float* __restrict__ za = xT + ((size_t)t * BATCH + row) * FREQ;
  const float* __restrict__ ha = h + (size_t)row * HID;
  const float* __restrict__ w0 = Wt + (size_t)(n0 + (lane & 15)) * ZDIM;
  const float* __restrict__ w1 = w0 + (size_t)16 * ZDIM;

  v8f acc0 = {}, acc1 = {};

  // K = 0..127 from x_t
  #pragma unroll 4
  for (int k0 = 0; k0 < FREQ; k0 += 4) {
    const int kk = k0 + khi;
    v2f a  = *(const v2f*)(za + kk);
    v2f b0 = *(const v2f*)(w0 + kk);
    v2f b1 = *(const v2f*)(w1 + kk);
    acc0 = __builtin_amdgcn_wmma_f32_16x16x4_f32(false, a, false, b0,
                                                 (short)0, acc0, false, false);
    acc1 = __builtin_amdgcn_wmma_f32_16x16x4_f32(false, a, false, b1,
                                                 (short)0, acc1, false, false);
  }
  // K = 128..1151 from h
  #pragma unroll 4
  for (int k0 = FREQ; k0 < ZDIM; k0 += 4) {
    const int kk = k0 + khi;
    v2f a  = *(const v2f*)(ha + (kk - FREQ));
    v2f b0 = *(const v2f*)(w0 + kk);
    v2f b1 = *(const v2f*)(w1 + kk);
    acc0 = __builtin_amdgcn_wmma_f32_16x16x4_f32(false, a, false, b0,
                                                 (short)0, acc0, false, false);
    acc1 = __builtin_amdgcn_wmma_f32_16x16x4_f32(false, a, false, b1,
                                                 (short)0, acc1, false, false);
  }

  // C/D layout: VGPR r, lanes 0-15 -> M=r, lanes 16-31 -> M=r+8; N = lane%16
  const int rbase = m * 16 + ((lane >> 4) << 3);
  const int nc    = n0 + (lane & 15);
  #pragma unroll
  for (int r = 0; r < 8; r++) {
    gates[(size_t)(rbase + r) * G4 + nc]      = acc0[r];
    gates[(size_t)(rbase + r) * G4 + nc + 16] = acc1[r];
  }
}

// ---------------------------------------------------------------------------
// block-wide sum over 256 threads (8 wave32 waves), broadcast to all threads
// ---------------------------------------------------------------------------
__device__ __forceinline__ float block_sum(float v, float* sm) {
  #pragma unroll
  for (int off = 16; off > 0; off >>= 1) v += __shfl_xor(v, off, 32);
  const int wv = threadIdx.x >> 5;
  const int ln = threadIdx.x & 31;
  __syncthreads();
  if (ln == 0) sm[wv] = v;
  __syncthreads();
  float s = 0.0f;
  #pragma unroll
  for (int w = 0; w < 8; w++) s += sm[w];
  return s;
}

// ---------------------------------------------------------------------------
// Per-step epilogue: gate nonlinearities, cell/hidden update, 2x layernorm,
// out-projection [1024]x[1024,8]. One block per batch row, 256 thr x 4 elem.
// ---------------------------------------------------------------------------
__global__ __launch_bounds__(256) void lstm_step_kernel(
    const float* __restrict__ gates,
    const float* __restrict__ bf, const float* __restrict__ bi,
    const float* __restrict__ bc, const float* __restrict__ bo,
    const float* __restrict__ ln_c_g, const float* __restrict__ ln_c_b,
    const float* __restrict__ ln_h_g, const float* __restrict__ ln_h_b,
    const float* __restrict__ W_out, const float* __restrict__ b_out,
    float* __restrict__ h, float* __restrict__ c,
    float* __restrict__ outp, int t) {
  __shared__ float sm[8];
  const int b   = blockIdx.x;
  const int tid = threadIdx.x;
  const float* __restrict__ g_row = gates + (size_t)b * G4;

  float cn[4], hn[4];
  float s_c = 0.f, q_c = 0.f, s_h = 0.f, q_h = 0.f;
  #pragma unroll
  for (int u = 0; u < 4; u++) {
    const int j = tid + u * 256;
    const float fg = sigm(g_row[j] + bf[j]);
    const float ig = sigm(g_row[HID + j] + bi[j]);
    const float gg = tanhf(g_row[2 * HID + j] + bc[j]);
    const float og = sigm(g_row[3 * HID + j] + bo[j]);
    const float cv = c[(size_t)b * HID + j] * fg + ig * gg;
    const float hv = og * tanhf(cv);
    cn[u] = cv; hn[u] = hv;
    s_c += cv; q_c += cv * cv;
    s_h += hv; q_h += hv * hv;
  }
  s_c = block_sum(s_c, sm);
  q_c = block_sum(q_c, sm);
  s_h = block_sum(s_h, sm);
  q_h = block_sum(q_h, sm);

  const float inv  = 1.0f / (float)HID;
  const float mu_c = s_c * inv;
  const float mu_h = s_h * inv;
  const float rs_c = rsqrtf(q_c * inv - mu_c * mu_c + LN_EPS);
  const float rs_h = rsqrtf(q_h * inv - mu_h * mu_h + LN_EPS);

  float oa[OUTD];
  #pragma unroll
  for (int o = 0; o < OUTD; o++) oa[o] = 0.0f;

  #pragma unroll
  for (int u = 0; u < 4; u++) {
    const int j = tid + u * 256;
    const float cl = (cn[u] - mu_c) * rs_c * ln_c_g[j] + ln_c_b[j];
    const float hl = (hn[u] - mu_h) * rs_h * ln_h_g[j] + ln_h_b[j];
    c[(size_t)b * HID + j] = cl;
    h[(size_t)b * HID + j] = hl;
    #pragma unroll
    for (int o = 0; o < OUTD; o++) oa[o] += hl * W_out[(size_t)j * OUTD + o];
  }

  float res[OUTD];
  #pragma unroll
  for (int o = 0; o < OUTD; o++) res[o] = block_sum(oa[o], sm);

  if (tid == 0) {
    #pragma unroll
    for (int o = 0; o < OUTD; o++)
      outp[(size_t)b * OUTD * TSEQ + (size_t)o * TSEQ + t] = res[o] + b_out[o];
  }
}

// ---------------------------------------------------------------------------
extern "C" void kernel_launch(void* const* d_in, const int* in_sizes, int n_in,
                              void* d_out, int out_size, void* d_ws, size_t ws_size,
                              hipStream_t stream) {
  (void)in_sizes; (void)n_in; (void)out_size; (void)ws_size;

  const float* x     = (const float*)d_in[0];
  const float* Wf    = (const float*)d_in[1];
  const float* bf    = (const float*)d_in[2];
  const float* Wi    = (const float*)d_in[3];
  const float* bi    = (const float*)d_in[4];
  const float* Wc    = (const float*)d_in[5];
  const float* bc    = (const float*)d_in[6];
  const float* Wo    = (const float*)d_in[7];
  const float* bo    = (const float*)d_in[8];
  const float* lncg  = (const float*)d_in[9];
  const float* lncb  = (const float*)d_in[10];
  const float* lnhg  = (const float*)d_in[11];
  const float* lnhb  = (const float*)d_in[12];
  const float* W_out = (const float*)d_in[13];
  const float* b_out = (const float*)d_in[14];

  float* ws    = (float*)d_ws;
  float* xT    = ws;                                   // T*B*FREQ   = 2,097,152
  float* Wt    = xT + (size_t)TSEQ * BATCH * FREQ;     // 4096*1152  = 4,718,592
  float* h     = Wt + (size_t)G4 * ZDIM;               // 65,536
  float* c     = h + (size_t)BATCH * HID;              // 65,536
  float* gates = c + (size_t)BATCH * HID;              // 262,144

  float* out   = (float*)d_out;                        // [B, OUT, T]
  float* h_out = out + (size_t)BATCH * OUTD * TSEQ;
  float* c_out = h_out + (size_t)BATCH * HID;

  transpose_x_kernel<<<(TSEQ * BATCH * FREQ) / 256, 256, 0, stream>>>(x, xT);
  transpose_w_kernel<<<dim3(HID / 32, ZDIM / 32, 4), 256, 0, stream>>>(Wf, Wi, Wc, Wo, Wt);
  zero_kernel<<<(2 * BATCH * HID) / 256, 256, 0, stream>>>(h, 2 * BATCH * HID);

  dim3 ggrid(4, 16);
  for (int t = 0; t < TSEQ; t++) {
    gate_gemm_kernel<<<ggrid, 256, 0, stream>>>(xT, h, Wt, gates, t);
    lstm_step_kernel<<<BATCH, 256, 0, stream>>>(gates, bf, bi, bc, bo,
                                                lncg, lncb, lnhg, lnhb,
                                                W_out, b_out, h, c, out, t);
  }

  hipMemcpyAsync(h_out, h, (size_t)BATCH * HID * sizeof(float),
                 hipMemcpyDeviceToDevice, stream);
  hipMemcpyAsync(c_out, c, (size_t)BATCH * HID * sizeof(float),
                 hipMemcpyDeviceToDevice, stream);
}